// SupConLossWithQueue2_8598524526702
// MI455X (gfx1250) — compile-verified
//
#include <hip/hip_runtime.h>
#include <hip/hip_bf16.h>

typedef float v2f __attribute__((ext_vector_type(2)));
typedef float v8f __attribute__((ext_vector_type(8)));

#define BB 1024
#define DD 128
#define QQ 65536
#define TEMP_INV 14.2857142857142857f /* 1/0.07 */

// ---------------------------------------------------------------------------
// Phase 0: zero the per-row accumulators (S_all, S_posexp, S_poslogit, count)
// ---------------------------------------------------------------------------
__global__ void supcon_zero_kernel(float* __restrict__ ws) {
    int i = blockIdx.x * blockDim.x + threadIdx.x;
    if (i < 4 * BB) ws[i] = 0.0f;
}

// ---------------------------------------------------------------------------
// Phase 1: fused GEMM (f2 @ queue_new^T) + masked row reductions.
// grid = (32 col chunks, 64 row blocks), block = 128 threads (4 waves).
// Each wave: 16 rows x 512 cols strip = 32 tiles of 16x16, K=128 via
// 32x v_wmma_f32_16x16x4_f32 per tile, split over 4 independent accumulator
// chains so the matrix pipe never stalls on its own previous result.
// ---------------------------------------------------------------------------
__global__ __launch_bounds__(128) void supcon_matmul_kernel(
    const float* __restrict__ features,   // (B, 2, 128): f1 = [:,0,:], f2 = [:,1,:]
    const int*   __restrict__ labels,     // (B,)
    const float* __restrict__ queue,      // (Q, 128)
    const int*   __restrict__ qlabels,    // (Q,)
    float*       __restrict__ ws)         // [0:1024)=S_all [1024:2048)=S_posexp
                                          // [2048:3072)=S_poslogit [3072:4096)=cnt
                                          // [4096:5120)=diag logits
{
    const int lane  = threadIdx.x & 31;
    const int wave  = threadIdx.x >> 5;
    const int n15   = lane & 15;
    const bool hi   = lane >= 16;         // upper 16-lane half of the wave
    const int rowBase  = blockIdx.y * 16;
    const int chunkCol = blockIdx.x * 2048;
    const int tile0    = wave * 32;       // 32 tiles (512 cols) per wave

    // --- A fragments: f2 row (rowBase + n15), all 128 K values, 64 VGPRs ---
    // A 16x4 f32 layout: lane%16 = M; VGPR0/1 hold K,K+1 (lanes 0-15) or
    // K+2,K+3 (lanes 16-31) -> contiguous float2 per lane per k-chunk.
    const float* aRow = features + (size_t)(rowBase + n15) * (2 * DD) + DD + (hi ? 2 : 0);
    v2f a[32];
#pragma unroll
    for (int k = 0; k < 32; ++k)
        a[k] = *(const v2f*)(aRow + 4 * k);

    // labels for the 8 rows this lane's C registers cover
    int labR[8];
#pragma unroll
    for (int r = 0; r < 8; ++r)
        labR[r] = labels[rowBase + r + (hi ? 8 : 0)];

    float sAll[8], sPos[8], sL[8], cnt[8];
#pragma unroll
    for (int r = 0; r < 8; ++r) { sAll[r] = 0.f; sPos[r] = 0.f; sL[r] = 0.f; cnt[r] = 0.f; }

    float* Diag = ws + 4 * BB;

    for (int t = 0; t < 32; ++t) {
        const int col0 = chunkCol + (tile0 + t) * 16;

        // queue_new column source: cols < B come from f1 (stride 256),
        // else from queue (stride 128). Tiles are 16-aligned, B%16==0,
        // so a tile never straddles the boundary.
        const float* qbase;
        int qstride, qlab;
        if (col0 < BB) {
            qbase   = features + (size_t)col0 * (2 * DD);
            qstride = 2 * DD;
            qlab    = labels[col0 + n15];
        } else {
            qbase   = queue + (size_t)(col0 - BB) * DD;
            qstride = DD;
            qlab    = qlabels[col0 - BB + n15];
        }
        // B 4x16 f32 layout (symmetric to A): lane%16 = N; VGPR0/1 = K,K+1
        // (lanes 0-15) or K+2,K+3 (lanes 16-31) -> contiguous float2.
        const float* colPtr = qbase + (size_t)n15 * qstride + (hi ? 2 : 0);

        // 4 independent accumulator chains -> pipelined matrix ops
        v8f c0 = {0.f,0.f,0.f,0.f,0.f,0.f,0.f,0.f};
        v8f c1 = {0.f,0.f,0.f,0.f,0.f,0.f,0.f,0.f};
        v8f c2 = {0.f,0.f,0.f,0.f,0.f,0.f,0.f,0.f};
        v8f c3 = {0.f,0.f,0.f,0.f,0.f,0.f,0.f,0.f};
#pragma unroll
        for (int k = 0; k < 32; k += 4) {
            v2f b0 = *(const v2f*)(colPtr + 4 * (k + 0));
            v2f b1 = *(const v2f*)(colPtr + 4 * (k + 1));
            v2f b2 = *(const v2f*)(colPtr + 4 * (k + 2));
            v2f b3 = *(const v2f*)(colPtr + 4 * (k + 3));
            c0 = __builtin_amdgcn_wmma_f32_16x16x4_f32(
                    false, a[k + 0], false, b0, (short)0, c0, false, false);
            c1 = __builtin_amdgcn_wmma_f32_16x16x4_f32(
                    false, a[k + 1], false, b1, (short)0, c1, false, false);
            c2 = __builtin_amdgcn_wmma_f32_16x16x4_f32(
                    false, a[k + 2], false, b2, (short)0, c2, false, false);
            c3 = __builtin_amdgcn_wmma_f32_16x16x4_f32(
                    false, a[k + 3], false, b3, (short)0, c3, false, false);
        }
        v8f c = (c0 + c1) + (c2 + c3);

        // C/D layout: lanes 0-15: N=lane, M=r ; lanes 16-31: N=lane-16, M=r+8
#pragma unroll
        for (int r = 0; r < 8; ++r) {
            float l = c[r] * TEMP_INV;        // logits (max-shift cancels algebraically)
            float e = __expf(l);
            bool pos = (labR[r] == qlab);
            sAll[r] += e;
            sPos[r] += pos ? e : 0.f;
            sL[r]   += pos ? l : 0.f;
            cnt[r]  += pos ? 1.f : 0.f;
            int rowg = rowBase + r + (hi ? 8 : 0);
            int colg = col0 + n15;
            if (rowg == colg) Diag[rowg] = l; // single writer per row per launch
        }
    }

    // Reduce across the 16 lanes of each half (xor masks 1,2,4,8 stay in-half).
#pragma unroll
    for (int r = 0; r < 8; ++r) {
#pragma unroll
        for (int m = 1; m < 16; m <<= 1) {
            sAll[r] += __shfl_xor(sAll[r], m, 32);
            sPos[r] += __shfl_xor(sPos[r], m, 32);
            sL[r]   += __shfl_xor(sL[r],   m, 32);
            cnt[r]  += __shfl_xor(cnt[r],  m, 32);
        }
    }
    if (n15 == 0) {   // lane 0 -> rows M=0..7, lane 16 -> rows M=8..15
        int roff = rowBase + (hi ? 8 : 0);
#pragma unroll
        for (int r = 0; r < 8; ++r) {
            atomicAdd(&ws[0 * BB + roff + r], sAll[r]);
            atomicAdd(&ws[1 * BB + roff + r], sPos[r]);
            atomicAdd(&ws[2 * BB + roff + r], sL[r]);
            atomicAdd(&ws[3 * BB + roff + r], cnt[r]);
        }
    }
}

// ---------------------------------------------------------------------------
// Phase 2: per-row losses + mean reduction -> (loss, self_loss, class_loss)
// ---------------------------------------------------------------------------
__global__ __launch_bounds__(1024) void supcon_finalize_kernel(
    const float* __restrict__ ws, float* __restrict__ out)
{
    const int i = threadIdx.x;
    float Sall = ws[0 * BB + i];
    float Spos = ws[1 * BB + i];
    float SL   = ws[2 * BB + i];
    float cntf = ws[3 * BB + i];
    float lii  = ws[4 * BB + i];

    float eii     = __expf(lii);
    float triplet = lii - __logf(Sall - eii);          // log_self_diff diagonal
    float logneg  = __logf(Sall - Spos);               // log sum(e * neg_mask)
    float cm      = cntf - 1.0f;                       // pos_mask excludes diagonal
    float mlpp    = (SL - lii) / cm - logneg;          // mean_log_prob_pos

    const float scale = 0.07f / 0.07f;                 // TEMP / BASE_TEMP
    float t = -scale * triplet;
    float c = -scale * mlpp;

    __shared__ float sT[1024];
    __shared__ float sC[1024];
    sT[i] = t; sC[i] = c;
    __syncthreads();
#pragma unroll
    for (int s = 512; s > 0; s >>= 1) {
        if (i < s) { sT[i] += sT[i + s]; sC[i] += sC[i + s]; }
        __syncthreads();
    }
    if (i == 0) {
        float mT = sT[0] * (1.0f / BB);   // self_loss
        float mC = sC[0] * (1.0f / BB);   // class_loss
        out[0] = 0.5f * (mT + mC);        // loss
        out[1] = mT;
        out[2] = mC;
    }
}

// ---------------------------------------------------------------------------
extern "C" void kernel_launch(void* const* d_in, const int* in_sizes, int n_in,
                              void* d_out, int out_size, void* d_ws, size_t ws_size,
                              hipStream_t stream) {
    const float* features = (const float*)d_in[0];  // (1024, 2, 128) f32
    const int*   labels   = (const int*)  d_in[1];  // (1024,) i32
    const float* queue    = (const float*)d_in[2];  // (65536, 128) f32
    const int*   qlabels  = (const int*)  d_in[3];  // (65536,) i32
    float* out = (float*)d_out;                      // 3 scalars
    float* ws  = (float*)d_ws;                       // needs 5120 floats = 20 KB

    supcon_zero_kernel<<<16, 256, 0, stream>>>(ws);
    supcon_matmul_kernel<<<dim3(32, 64), 128, 0, stream>>>(
        features, labels, queue, qlabels, ws);
    supcon_finalize_kernel<<<1, 1024, 0, stream>>>(ws, out);
}